// PointcloudToVoxels_41704132444291
// MI455X (gfx1250) — compile-verified
//
#include <hip/hip_runtime.h>
#include <stdint.h>

// ---------------- problem constants (match reference) ----------------
#define WGRID 64
#define LGRID 64
#define HGRID 64
#define NVOX  (WGRID * LGRID * HGRID)      // 262144
#define NBATCH 4
#define NCHAN  32
#define NPTS   200000
#define VOXSZ  0.05f
#define MINPTS 10

#define GLOBAL_AS __attribute__((address_space(1)))
#define LDS_AS    __attribute__((address_space(3)))

#if defined(__has_builtin)
#if __has_builtin(__builtin_amdgcn_global_load_async_to_lds_b32)
#define HAVE_ASYNC_LDS_BUILTIN 1
#endif
#if __has_builtin(__builtin_amdgcn_s_wait_asynccnt)
#define HAVE_WAIT_ASYNC_BUILTIN 1
#endif
#endif

// ---- order-preserving float<->u32 mapping so MAX can use native u32 atomics
__device__ __forceinline__ unsigned f2ord(float f) {
    unsigned u = __float_as_uint(f);
    return (u & 0x80000000u) ? ~u : (u | 0x80000000u);
}
__device__ __forceinline__ float ord2f(unsigned u) {
    unsigned b = (u & 0x80000000u) ? (u & 0x7fffffffu) : ~u;
    return __uint_as_float(b);
}

__device__ __forceinline__ void async_ld_f32_to_lds(const float* gptr, float* lptr) {
#if defined(HAVE_ASYNC_LDS_BUILTIN)
    // clang signature: (int addrspace(1)*, int addrspace(3)*, imm offset, imm cpol)
    __builtin_amdgcn_global_load_async_to_lds_b32(
        (GLOBAL_AS int*)(void*)gptr, (LDS_AS int*)(void*)lptr, 0, 0);
#else
    unsigned lds_off = (unsigned)(uintptr_t)(LDS_AS void*)lptr;
    unsigned long long ga = (unsigned long long)(uintptr_t)gptr;
    asm volatile("global_load_async_to_lds_b32 %0, %1, off"
                 :: "v"(lds_off), "v"(ga)
                 : "memory");
#endif
}

__device__ __forceinline__ void wait_asynccnt0() {
#if defined(HAVE_WAIT_ASYNC_BUILTIN)
    __builtin_amdgcn_s_wait_asynccnt(0);
#else
    asm volatile("s_wait_asynccnt 0x0" ::: "memory");
#endif
}

// ---------------- kernel 1: zero-fill output (uint4 stores) ----------------
__global__ __launch_bounds__(256) void vox_init(uint4* __restrict__ out, int n4) {
    int i = blockIdx.x * blockDim.x + threadIdx.x;
    if (i < n4) {
        uint4 z; z.x = 0u; z.y = 0u; z.z = 0u; z.w = 0u;
        out[i] = z;
    }
}

// ---------------- kernel 2: scatter (count + ordered-u32 max) ----------------
__global__ __launch_bounds__(256) void vox_scatter(const float* __restrict__ coords,  // [B,3,N]
                                                   const float* __restrict__ attrs,   // [B,C,N]
                                                   unsigned*    __restrict__ data_u,  // [B,C,V]
                                                   unsigned*    __restrict__ cnt_u) { // [B,V]
    __shared__ float sc[3 * 256];
    const int tid = threadIdx.x;
    const int blocksPerBatch = (NPTS + 255) / 256;
    const int b = blockIdx.x / blocksPerBatch;
    const int n = (blockIdx.x % blocksPerBatch) * 256 + tid;
    const bool valid = (n < NPTS);

    const float* cb = coords + (size_t)b * 3 * NPTS;
    const float* ab = attrs  + (size_t)b * NCHAN * NPTS + n;

    // Stage the three coordinate streams through the CDNA5 async-to-LDS path.
    if (valid) {
        async_ld_f32_to_lds(cb + n,             &sc[tid]);
        async_ld_f32_to_lds(cb + NPTS + n,      &sc[256 + tid]);
        async_ld_f32_to_lds(cb + 2 * NPTS + n,  &sc[512 + tid]);
        // Warm the attribute stream while the async copy is in flight.
        __builtin_prefetch(ab, 0, 0);
    }
    wait_asynccnt0();  // each lane reads back only its own staged slots

    if (!valid) return;

    const float x = sc[tid];
    const float y = sc[256 + tid];
    const float z = sc[512 + tid];

    // exact reference formula: floor((p - 0)/0.05 + 0.5), then int cast
    const int ix = (int)floorf(x / VOXSZ + 0.5f);
    const int iy = (int)floorf(y / VOXSZ + 0.5f);
    const int iz = (int)floorf(z / VOXSZ + 0.5f);
    if ((unsigned)ix >= (unsigned)WGRID ||
        (unsigned)iy >= (unsigned)LGRID ||
        (unsigned)iz >= (unsigned)HGRID) return;

    const int flat = ix * (LGRID * HGRID) + iy * HGRID + iz;

    atomicAdd(&cnt_u[(size_t)b * NVOX + flat], 1u);

    unsigned* db = data_u + (size_t)b * NCHAN * NVOX + flat;
#pragma unroll
    for (int c = 0; c < NCHAN; ++c) {
        const float v = ab[(size_t)c * NPTS];           // coalesced across lanes
        atomicMax(db + (size_t)c * NVOX, f2ord(v));     // native GLOBAL_ATOMIC_MAX_U32
    }
}

// ---------------- kernel 3: finalize, 4 voxels/thread with b128 traffic ----------------
__global__ __launch_bounds__(256) void vox_finalize(unsigned* __restrict__ data_u,  // [B,C,V] in-place
                                                    unsigned* __restrict__ cnt_u) { // [B,V] -> occ float
    const int NQ = NVOX / 4;                               // voxel quads per batch
    const int i = blockIdx.x * blockDim.x + threadIdx.x;   // over B*NQ
    if (i >= NBATCH * NQ) return;
    const int b  = i / NQ;
    const int vq = i - b * NQ;                             // quad index within batch
    const size_t v = (size_t)vq * 4;

    const uint4 cnt = *(const uint4*)(cnt_u + (size_t)b * NVOX + v);

    const size_t base = (size_t)b * NCHAN * NVOX + v;
#pragma unroll
    for (int c = 0; c < NCHAN; ++c) {
        const size_t idx = base + (size_t)c * NVOX;
        const uint4 u = *(const uint4*)(data_u + idx);
        float4 f;
        f.x = (cnt.x > 0u) ? ord2f(u.x) : 0.0f;
        f.y = (cnt.y > 0u) ? ord2f(u.y) : 0.0f;
        f.z = (cnt.z > 0u) ? ord2f(u.z) : 0.0f;
        f.w = (cnt.w > 0u) ? ord2f(u.w) : 0.0f;
        *(float4*)(data_u + idx) = f;                      // in-place u32 -> f32
    }

    float4 occ;
    occ.x = (cnt.x >= (unsigned)MINPTS) ? 1.0f : 0.0f;
    occ.y = (cnt.y >= (unsigned)MINPTS) ? 1.0f : 0.0f;
    occ.z = (cnt.z >= (unsigned)MINPTS) ? 1.0f : 0.0f;
    occ.w = (cnt.w >= (unsigned)MINPTS) ? 1.0f : 0.0f;
    *(float4*)(cnt_u + (size_t)b * NVOX + v) = occ;        // in-place count -> occupancy
}

// ---------------- launch ----------------
extern "C" void kernel_launch(void* const* d_in, const int* in_sizes, int n_in,
                              void* d_out, int out_size, void* d_ws, size_t ws_size,
                              hipStream_t stream) {
    const float* coords = (const float*)d_in[0];   // [4,3,200000]
    const float* attrs  = (const float*)d_in[1];   // [4,32,200000]

    unsigned* data_u = (unsigned*)d_out;                       // [B,C,V]
    unsigned* cnt_u  = data_u + (size_t)NBATCH * NCHAN * NVOX; // [B,V] (occupancy region)

    // 1) zero output (counts + ordered-max identities) with 128-bit stores
    const int n4 = out_size / 4;  // out_size divisible by 4
    vox_init<<<(n4 + 255) / 256, 256, 0, stream>>>((uint4*)d_out, n4);

    // 2) scatter points
    const int blocksPerBatch = (NPTS + 255) / 256;
    vox_scatter<<<NBATCH * blocksPerBatch, 256, 0, stream>>>(coords, attrs, data_u, cnt_u);

    // 3) finalize (4 voxels per thread, uint4/float4 accesses)
    const int nq = NBATCH * (NVOX / 4);
    vox_finalize<<<(nq + 255) / 256, 256, 0, stream>>>(data_u, cnt_u);
}